// GNN_42236708388990
// MI455X (gfx1250) — compile-verified
//
#include <hip/hip_runtime.h>
#include <hip/hip_bf16.h>
#include <stddef.h>

// Problem constants (from reference): B=8, N=2048, D=64, H=128, DEG=16
#define GB   8
#define GN   2048
#define GD   64
#define GH   128
#define GDEG 16
#define GM   (GB * GN)            // 16384 rows
#define PLANE ((size_t)GM * GH)   // elements in one [M,H] buffer = 2,097,152
// LDS pair-row stride in floats (dwords). 288 % 64 == 32 -> the two half-wave
// pair-rows of a WMMA B-fragment b64 load cover disjoint 32-bank sets.
#define PSTR 288

typedef __attribute__((ext_vector_type(2))) float v2f;
typedef __attribute__((ext_vector_type(8))) float v8f;

// Stage rows [0,K) of W[.,128] into LDS as interleaved K-pair rows:
// wlds[p*PSTR + 2n .. 2n+1] = { W[2p, n], W[2p+1, n] }
template <int K>
__device__ __forceinline__ void stage_w_pairs(const float* __restrict__ W,
                                              float* __restrict__ wlds, int tid)
{
#pragma unroll
  for (int i = tid; i < (K / 2) * 32; i += 256) {
    const int p = i >> 5;
    const int c = (i & 31) << 2;
    const float4 a0 = *reinterpret_cast<const float4*>(W + (size_t)(2 * p) * GH + c);
    const float4 a1 = *reinterpret_cast<const float4*>(W + (size_t)(2 * p + 1) * GH + c);
    float2* dst = reinterpret_cast<float2*>(&wlds[p * PSTR + 2 * c]);
    float2 d0; d0.x = a0.x; d0.y = a1.x; dst[0] = d0;
    float2 d1; d1.x = a0.y; d1.y = a1.y; dst[1] = d1;
    float2 d2; d2.x = a0.z; d2.y = a1.z; dst[2] = d2;
    float2 d3; d3.x = a0.w; d3.y = a1.w; dst[3] = d3;
  }
}

// ---------------------------------------------------------------------------
// Encoder GEMM + bias + ReLU:  Y[M,128] = relu(X[M,K] @ W[K,128] + b)
// W staged in LDS (pair-interleaved) once per block; each wave computes a
// 16-row x 128-col stripe as 8 WMMA 16x16 tiles, K-loop in steps of 4.
// blockDim = 256 (8 waves), gridDim.x = M/128 = 128.
// ---------------------------------------------------------------------------
template <int K>
__global__ __launch_bounds__(256) void enc_gemm_relu(
    const float* __restrict__ X, const float* __restrict__ W,
    const float* __restrict__ bias, float* __restrict__ Y)
{
  __shared__ float wlds[(K / 2) * PSTR];

  const int tid = threadIdx.x;
  stage_w_pairs<K>(W, wlds, tid);
  __syncthreads();

  const int lane = tid & 31;
  const int half = lane >> 4;       // selects K pair (k,k+1) vs (k+2,k+3)
  const int l16  = lane & 15;
  const int wave = tid >> 5;
  const int m0   = (blockIdx.x * 8 + wave) * 16;

  const float* __restrict__ xrow = X + (size_t)(m0 + l16) * (size_t)K;

  const v8f vzero = {0.f, 0.f, 0.f, 0.f, 0.f, 0.f, 0.f, 0.f};
  v8f acc[8];
#pragma unroll
  for (int t = 0; t < 8; ++t) acc[t] = vzero;

#pragma unroll 4
  for (int k = 0; k < K; k += 4) {
    const int ka = k + half * 2;
    v2f a;
    a.x = xrow[ka];
    a.y = xrow[ka + 1];
    const float* prow = &wlds[((k >> 1) + half) * PSTR];
#pragma unroll
    for (int t = 0; t < 8; ++t) {
      const int n = t * 16 + l16;
      const v2f bf = *reinterpret_cast<const v2f*>(&prow[2 * n]);  // ds_load_b64
      acc[t] = __builtin_amdgcn_wmma_f32_16x16x4_f32(
          false, a, false, bf, (short)0, acc[t], false, false);
    }
  }

  // C/D layout: VGPR r -> row m0 + half*8 + r, col = tile*16 + l16
#pragma unroll
  for (int t = 0; t < 8; ++t) {
    const int col = t * 16 + l16;
    const float bv = bias[col];
#pragma unroll
    for (int r = 0; r < 8; ++r) {
      const int row = m0 + half * 8 + r;
      float v = acc[t][r] + bv;
      Y[(size_t)row * GH + col] = fmaxf(v, 0.f);
    }
  }
}

// ---------------------------------------------------------------------------
// Update GEMM: hu[M,128] = relu(concat(agg, ie)[M,256] @ Wu[256,128] + bu)
// Wu (128KB) staged into LDS in two pair-interleaved halves; the half swap
// coincides with the concat boundary (AGG rows then IE rows).
// ---------------------------------------------------------------------------
__global__ __launch_bounds__(256) void upd_gemm_relu(
    const float* __restrict__ AGG, const float* __restrict__ IE,
    const float* __restrict__ Wu, const float* __restrict__ bias,
    float* __restrict__ Y)
{
  __shared__ float wlds[(GH / 2) * PSTR];

  const int tid  = threadIdx.x;
  const int lane = tid & 31;
  const int half = lane >> 4;
  const int l16  = lane & 15;
  const int wave = tid >> 5;
  const int m0   = (blockIdx.x * 8 + wave) * 16;

  const size_t rowoff = (size_t)(m0 + l16) * GH;

  const v8f vzero = {0.f, 0.f, 0.f, 0.f, 0.f, 0.f, 0.f, 0.f};
  v8f acc[8];
#pragma unroll
  for (int t = 0; t < 8; ++t) acc[t] = vzero;

#pragma unroll
  for (int phase = 0; phase < 2; ++phase) {
    const float* __restrict__ Wsrc = Wu + (size_t)phase * GH * GH;
    const float* __restrict__ xb   = phase ? IE : AGG;

    if (phase) __syncthreads();  // waves done reading previous half
    stage_w_pairs<GH>(Wsrc, wlds, tid);
    __syncthreads();

#pragma unroll 4
    for (int k = 0; k < GH; k += 4) {
      const int ka = k + half * 2;
      v2f a;
      a.x = xb[rowoff + ka];
      a.y = xb[rowoff + ka + 1];
      const float* prow = &wlds[((k >> 1) + half) * PSTR];
#pragma unroll
      for (int t = 0; t < 8; ++t) {
        const int n = t * 16 + l16;
        const v2f bf = *reinterpret_cast<const v2f*>(&prow[2 * n]);
        acc[t] = __builtin_amdgcn_wmma_f32_16x16x4_f32(
            false, a, false, bf, (short)0, acc[t], false, false);
      }
    }
  }

#pragma unroll
  for (int t = 0; t < 8; ++t) {
    const int col = t * 16 + l16;
    const float bv = bias[col];
#pragma unroll
    for (int r = 0; r < 8; ++r) {
      const int row = m0 + half * 8 + r;
      float v = acc[t][r] + bv;
      Y[(size_t)row * GH + col] = fmaxf(v, 0.f);
    }
  }
}

// ---------------------------------------------------------------------------
// Aggregation: agg[b,n,h] = sum_d se[b, sn[n,d], h] + sum_d ie[b, in[n,d], h]
// One wave per (b,n) node; each lane owns 4 h's (float4) -> 512B/row/wave,
// fully coalesced, all gathered rows L2-resident.
// ---------------------------------------------------------------------------
__global__ __launch_bounds__(128) void aggregate_kernel(
    const float* __restrict__ se, const float* __restrict__ ie,
    const int* __restrict__ sn, const int* __restrict__ inn,
    float* __restrict__ agg)
{
  const int wave = threadIdx.x >> 5;
  const int lane = threadIdx.x & 31;
  const int bn = blockIdx.x * 4 + wave;      // 0 .. B*N-1
  const int b  = bn >> 11;                   // / N (N = 2048)
  const int n  = bn & (GN - 1);

  const float4* __restrict__ seb = reinterpret_cast<const float4*>(se) + (size_t)b * GN * 32;
  const float4* __restrict__ ieb = reinterpret_cast<const float4*>(ie) + (size_t)b * GN * 32;

  float sx = 0.f, sy = 0.f, sz = 0.f, sw = 0.f;
#pragma unroll
  for (int d = 0; d < GDEG; ++d) {
    const int j = sn[n * GDEG + d];
    const float4 v = seb[(size_t)j * 32 + lane];
    sx += v.x; sy += v.y; sz += v.z; sw += v.w;
  }
#pragma unroll
  for (int d = 0; d < GDEG; ++d) {
    const int j = inn[n * GDEG + d];
    const float4 v = ieb[(size_t)j * 32 + lane];
    sx += v.x; sy += v.y; sz += v.z; sw += v.w;
  }
  float4 r; r.x = sx; r.y = sy; r.z = sz; r.w = sw;
  reinterpret_cast<float4*>(agg)[(size_t)bn * 32 + lane] = r;
}

// ---------------------------------------------------------------------------
extern "C" void kernel_launch(void* const* d_in, const int* in_sizes, int n_in,
                              void* d_out, int out_size, void* d_ws, size_t ws_size,
                              hipStream_t stream)
{
  const float* state     = (const float*)d_in[0];
  const float* internal_ = (const float*)d_in[1];
  const int*   sn        = (const int*)d_in[2];
  const int*   inn       = (const int*)d_in[3];
  const float* W1 = (const float*)d_in[4];   const float* b1 = (const float*)d_in[5];
  const float* W2 = (const float*)d_in[6];   const float* b2 = (const float*)d_in[7];
  const float* W3 = (const float*)d_in[8];   const float* b3 = (const float*)d_in[9];
  const float* Wu1 = (const float*)d_in[10]; const float* bu1 = (const float*)d_in[11];
  const float* Wu2 = (const float*)d_in[12]; const float* bu2 = (const float*)d_in[13];
  const float* Wu3 = (const float*)d_in[14]; const float* bu3 = (const float*)d_in[15];

  float* out = (float*)d_out;
  float* hu1 = out;
  float* hu2 = out + PLANE;
  float* hu3 = out + 2 * PLANE;

  float* ws   = (float*)d_ws;
  float* se_a = ws;
  float* se_b = ws + PLANE;
  float* ie   = ws + 2 * PLANE;
  float* agg  = ws + 3 * PLANE;

  const dim3 gg(GM / 128);   // 128 blocks
  const dim3 gb(256);        // 8 waves/block
  const dim3 ag(GM / 4);     // one wave per (b,n)
  const dim3 ab(128);

  // Layer 1
  enc_gemm_relu<GD><<<gg, gb, 0, stream>>>(state,     W1, b1, se_a);
  enc_gemm_relu<GD><<<gg, gb, 0, stream>>>(internal_, W1, b1, ie);
  aggregate_kernel<<<ag, ab, 0, stream>>>(se_a, ie, sn, inn, agg);
  upd_gemm_relu<<<gg, gb, 0, stream>>>(agg, ie, Wu1, bu1, hu1);

  // Layer 2
  enc_gemm_relu<GH><<<gg, gb, 0, stream>>>(se_a, W2, b2, se_b);
  enc_gemm_relu<GH><<<gg, gb, 0, stream>>>(hu1,  W2, b2, ie);
  aggregate_kernel<<<ag, ab, 0, stream>>>(se_b, ie, sn, inn, agg);
  upd_gemm_relu<<<gg, gb, 0, stream>>>(agg, ie, Wu2, bu2, hu2);

  // Layer 3
  enc_gemm_relu<GH><<<gg, gb, 0, stream>>>(se_b, W3, b3, se_a);
  enc_gemm_relu<GH><<<gg, gb, 0, stream>>>(hu2,  W3, b3, ie);
  aggregate_kernel<<<ag, ab, 0, stream>>>(se_a, ie, sn, inn, agg);
  upd_gemm_relu<<<gg, gb, 0, stream>>>(agg, ie, Wu3, bu3, hu3);
}